// DeepLagrangianNetwork_13417477832894
// MI455X (gfx1250) — compile-verified
//
#include <hip/hip_runtime.h>
#include <hip/hip_bf16.h>
#include <math.h>

// ---------------------------------------------------------------------------
// DeLaN fused kernel for MI455X (gfx1250, wave32, f32 WMMA 16x16x4)
// Dual-accumulator WMMA chains for XDL pipelining.
// ---------------------------------------------------------------------------

typedef __attribute__((ext_vector_type(2))) float v2f;
typedef __attribute__((ext_vector_type(8))) float v8f;

#define QD   12
#define HIDN 64
#define NLO  66
#define NS   16            // samples per wave
#define WPW  2             // waves per workgroup
#define SMP_PER_WG (NS * WPW)

// per-wave LDS floats: cos(192) sin(192) h1(1024) h2(1024) ho(1536) sig3(192) Kt(1024)
#define LDSF_WAVE 5184
#define SMEM_BYTES (WPW * LDSF_WAVE * 4)

__device__ __forceinline__ v8f vzero8() {
    v8f z = {0.f, 0.f, 0.f, 0.f, 0.f, 0.f, 0.f, 0.f};
    return z;
}

__device__ __forceinline__ v8f wmma4(v2f a, v2f b, v8f c) {
#if __has_builtin(__builtin_amdgcn_wmma_f32_16x16x4_f32)
    // (neg_a, A, neg_b, B, c_mod, C, reuse_a, reuse_b)
    return __builtin_amdgcn_wmma_f32_16x16x4_f32(false, a, false, b,
                                                 (short)0, c, false, false);
#else
    c[0] += a.x * b.x + a.y * b.y;   // degenerate fallback; histogram reveals
    return c;
#endif
}

__device__ __forceinline__ void wave_sync() {
#if __has_builtin(__builtin_amdgcn_s_wait_dscnt)
    __builtin_amdgcn_s_wait_dscnt(0);
#else
    asm volatile("s_wait_dscnt 0" ::: "memory");
#endif
    __builtin_amdgcn_wave_barrier();
}

__device__ __forceinline__ float leaky(float x) { return x > 0.f ? x : 0.01f * x; }

// ---------------------------------------------------------------------------
// Kernel 1: heavy WMMA pipeline. 2 waves / 64 threads per WG, 16 samples/wave.
// Writes per sample to workspace: L(144), g(12), y=L^T qdot(12),
// dld_dq(144 = [l][j]), dlo_dq(792 = [o][j]).
// ---------------------------------------------------------------------------
__global__ __launch_bounds__(64)
void delan_main(const float* __restrict__ xu,
                const float* __restrict__ W1, const float* __restrict__ b1,
                const float* __restrict__ W2, const float* __restrict__ b2,
                const float* __restrict__ WG, const float* __restrict__ bG,
                const float* __restrict__ WLd, const float* __restrict__ bLd,
                const float* __restrict__ WLo, const float* __restrict__ bLo,
                float* __restrict__ Lg, float* __restrict__ gg,
                float* __restrict__ yy, float* __restrict__ dld,
                float* __restrict__ dlo, int n)
{
    extern __shared__ float smem[];
    const int tid  = threadIdx.x;
    const int wv   = tid >> 5;
    const int lane = tid & 31;
    const int half = lane >> 4;
    const int l16  = lane & 15;

    float* wb    = smem + wv * LDSF_WAVE;
    float* cosq  = wb;            // 16 x 12
    float* sinq  = wb + 192;      // 16 x 12
    float* h1s   = wb + 384;      // 16 x 64 (post-activation)
    float* h2s   = wb + 1408;     // 16 x 64 (post-activation)
    float* ho    = wb + 2432;     // 16 x 96 (h3 | g | Lo | pad)
    float* sig3s = wb + 3968;     // 16 x 12
    float* Kt    = wb + 4160;     // 64 x 16 column tile of K

    const int base = blockIdx.x * SMP_PER_WG + wv * NS;

    // ---- trig precompute ---------------------------------------------------
    if (lane < NS) {
        const int gs = base + lane;
        #pragma unroll
        for (int j = 0; j < QD; ++j) {
            float q = xu[gs * 36 + j];
            cosq[lane * 12 + j] = cosf(q);
            sinq[lane * 12 + j] = sinf(q);
        }
    }
    wave_sync();

    // ---- GEMM1: h1 = leaky(emb @ W1^T + b1), M=16 K=24 N=64 ---------------
    for (int Nt = 0; Nt < 4; ++Nt) {
        v8f acc0 = vzero8(), acc1 = vzero8();
        const int ncol = Nt * 16 + l16;
        const float* w1r = W1 + ncol * 24;
        #pragma unroll
        for (int kc = 0; kc < 6; kc += 2) {
            {   // even chunk -> chain 0
                const int k0 = kc * 4 + 2 * half;
                v2f a, b;
                a.x = (k0     < 12) ? cosq[l16 * 12 + k0]     : sinq[l16 * 12 + k0 - 12];
                a.y = (k0 + 1 < 12) ? cosq[l16 * 12 + k0 + 1] : sinq[l16 * 12 + k0 - 11];
                b.x = w1r[k0];
                b.y = w1r[k0 + 1];
                acc0 = wmma4(a, b, acc0);
            }
            {   // odd chunk -> chain 1
                const int k0 = (kc + 1) * 4 + 2 * half;
                v2f a, b;
                a.x = (k0     < 12) ? cosq[l16 * 12 + k0]     : sinq[l16 * 12 + k0 - 12];
                a.y = (k0 + 1 < 12) ? cosq[l16 * 12 + k0 + 1] : sinq[l16 * 12 + k0 - 11];
                b.x = w1r[k0];
                b.y = w1r[k0 + 1];
                acc1 = wmma4(a, b, acc1);
            }
        }
        const float bias = b1[ncol];
        #pragma unroll
        for (int v = 0; v < 8; ++v) {
            const int m = v + 8 * half;
            h1s[m * 64 + ncol] = leaky(acc0[v] + acc1[v] + bias);
        }
    }
    wave_sync();

    // ---- GEMM2: h2 = leaky(h1 @ W2^T + b2), K=64 N=64 ---------------------
    for (int Nt = 0; Nt < 4; ++Nt) {
        v8f acc0 = vzero8(), acc1 = vzero8();
        const int ncol = Nt * 16 + l16;
        const float* w2r = W2 + ncol * 64;
        #pragma unroll
        for (int kc = 0; kc < 16; kc += 2) {
            {
                const int k0 = kc * 4 + 2 * half;
                v2f a, b;
                a.x = h1s[l16 * 64 + k0];
                a.y = h1s[l16 * 64 + k0 + 1];
                b.x = w2r[k0];
                b.y = w2r[k0 + 1];
                acc0 = wmma4(a, b, acc0);
            }
            {
                const int k0 = (kc + 1) * 4 + 2 * half;
                v2f a, b;
                a.x = h1s[l16 * 64 + k0];
                a.y = h1s[l16 * 64 + k0 + 1];
                b.x = w2r[k0];
                b.y = w2r[k0 + 1];
                acc1 = wmma4(a, b, acc1);
            }
        }
        const float bias = b2[ncol];
        #pragma unroll
        for (int v = 0; v < 8; ++v) {
            const int m = v + 8 * half;
            h2s[m * 64 + ncol] = leaky(acc0[v] + acc1[v] + bias);
        }
    }
    wave_sync();

    // ---- GEMM3: heads = h2 @ [WLd|WG|WLo]^T + b, N=90 (pad 96) ------------
    for (int Nt = 0; Nt < 6; ++Nt) {
        v8f acc0 = vzero8(), acc1 = vzero8();
        const int r2 = Nt * 16 + l16;
        const float* hw;
        float hb;
        bool valid = true;
        if (r2 < 12)      { hw = WLd + r2 * 64;        hb = bLd[r2]; }
        else if (r2 < 24) { hw = WG  + (r2 - 12) * 64; hb = bG[r2 - 12]; }
        else if (r2 < 90) { hw = WLo + (r2 - 24) * 64; hb = bLo[r2 - 24]; }
        else              { hw = WLd;                  hb = 0.f; valid = false; }
        #pragma unroll
        for (int kc = 0; kc < 16; kc += 2) {
            {
                const int k0 = kc * 4 + 2 * half;
                v2f a, b;
                a.x = h2s[l16 * 64 + k0];
                a.y = h2s[l16 * 64 + k0 + 1];
                b.x = valid ? hw[k0]     : 0.f;
                b.y = valid ? hw[k0 + 1] : 0.f;
                acc0 = wmma4(a, b, acc0);
            }
            {
                const int k0 = (kc + 1) * 4 + 2 * half;
                v2f a, b;
                a.x = h2s[l16 * 64 + k0];
                a.y = h2s[l16 * 64 + k0 + 1];
                b.x = valid ? hw[k0]     : 0.f;
                b.y = valid ? hw[k0 + 1] : 0.f;
                acc1 = wmma4(a, b, acc1);
            }
        }
        #pragma unroll
        for (int v = 0; v < 8; ++v) {
            const int m = v + 8 * half;
            ho[m * 96 + r2] = acc0[v] + acc1[v] + hb;
        }
    }
    wave_sync();

    // ---- per-sample forward epilogue: L, y, g, sig3 -----------------------
    if (lane < NS) {
        const int s  = lane;
        const int gs = base + s;
        float qd[12];
        #pragma unroll
        for (int j = 0; j < 12; ++j) qd[j] = xu[gs * 36 + 12 + j];

        float Ldg[12], yv[12];
        #pragma unroll
        for (int j = 0; j < 12; ++j) {
            float h3 = ho[s * 96 + j];
            sig3s[s * 12 + j] = 1.f / (1.f + expf(-h3));
            Ldg[j] = (h3 > 20.f) ? h3 : log1pf(expf(h3));   // softplus
            yv[j]  = Ldg[j] * qd[j];
        }
        float* Ls = Lg + (size_t)gs * 144;
        for (int i = 0; i < 144; ++i) Ls[i] = 0.f;
        #pragma unroll
        for (int j = 0; j < 12; ++j) Ls[j * 13] = Ldg[j];
        {
            int o = 0;
            for (int r = 1; r < 12; ++r)
                for (int c = 0; c < r; ++c, ++o) {
                    float lov = ho[s * 96 + 24 + o];
                    Ls[r * 12 + c] = lov;
                    yv[c] += lov * qd[r];
                }
        }
        #pragma unroll
        for (int j = 0; j < 12; ++j) {
            yy[gs * 12 + j] = yv[j];
            gg[gs * 12 + j] = ho[s * 96 + 12 + j];
        }
    }
    wave_sync();

    // ---- Jacobian GEMMs: 192 cols = 16 samples x 12 dirs ------------------
    for (int Nt2 = 0; Nt2 < 12; ++Nt2) {
        const int col = Nt2 * 16 + l16;
        const int sc  = col / 12;
        const int dir = col % 12;
        const float sq = sinq[sc * 12 + dir];
        const float cq = cosq[sc * 12 + dir];

        // Kpre = W2 @ dh1_dq (dh1 built on the fly), M=64 (4 tiles), K=64
        // 4 interleaved accumulator chains already give XDL ILP here.
        v8f accK[4];
        #pragma unroll
        for (int t = 0; t < 4; ++t) accK[t] = vzero8();
        #pragma unroll
        for (int kc = 0; kc < 16; ++kc) {
            const int k0 = kc * 4 + 2 * half;
            const float* w1a = W1 + k0 * 24;
            const float* w1b = w1a + 24;
            const float d0 = (h1s[sc * 64 + k0]     > 0.f) ? 1.f : -0.01f;
            const float d1 = (h1s[sc * 64 + k0 + 1] > 0.f) ? 1.f : -0.01f;
            v2f b;
            b.x = d0 * (w1a[12 + dir] * cq - w1a[dir] * sq);
            b.y = d1 * (w1b[12 + dir] * cq - w1b[dir] * sq);
            #pragma unroll
            for (int t = 0; t < 4; ++t) {
                const float* w2r = W2 + (t * 16 + l16) * 64 + k0;
                v2f a;
                a.x = w2r[0];
                a.y = w2r[1];
                accK[t] = wmma4(a, b, accK[t]);
            }
        }
        // apply dR2, stage K column-tile to LDS (row-major [g][c])
        #pragma unroll
        for (int t = 0; t < 4; ++t) {
            #pragma unroll
            for (int v = 0; v < 8; ++v) {
                const int g = t * 16 + v + 8 * half;
                const float dr2 = (h2s[sc * 64 + g] > 0.f) ? 1.f : -0.01f;
                Kt[g * 16 + l16] = accK[t][v] * dr2;
            }
        }
        wave_sync();

        // heads-deriv: [WLd;WLo](78x64) @ Ktile(64x16), M padded to 80
        for (int Mt2 = 0; Mt2 < 5; ++Mt2) {
            const int r = Mt2 * 16 + l16;
            const float* hw = (r < 12) ? (WLd + r * 64)
                             : (r < 78) ? (WLo + (r - 12) * 64) : WLd;
            const bool valid = (r < 78);
            v8f acc0 = vzero8(), acc1 = vzero8();
            #pragma unroll
            for (int kc2 = 0; kc2 < 16; kc2 += 2) {
                {
                    const int k0 = kc2 * 4 + 2 * half;
                    v2f a, b;
                    a.x = valid ? hw[k0]     : 0.f;
                    a.y = valid ? hw[k0 + 1] : 0.f;
                    b.x = Kt[k0 * 16 + l16];
                    b.y = Kt[(k0 + 1) * 16 + l16];
                    acc0 = wmma4(a, b, acc0);
                }
                {
                    const int k0 = (kc2 + 1) * 4 + 2 * half;
                    v2f a, b;
                    a.x = valid ? hw[k0]     : 0.f;
                    a.y = valid ? hw[k0 + 1] : 0.f;
                    b.x = Kt[k0 * 16 + l16];
                    b.y = Kt[(k0 + 1) * 16 + l16];
                    acc1 = wmma4(a, b, acc1);
                }
            }
            const int gs = base + sc;
            #pragma unroll
            for (int v = 0; v < 8; ++v) {
                const int r2 = Mt2 * 16 + v + 8 * half;
                const float val = acc0[v] + acc1[v];
                if (r2 < 12)
                    dld[(size_t)gs * 144 + r2 * 12 + dir] = sig3s[sc * 12 + r2] * val;
                else if (r2 < 78)
                    dlo[(size_t)gs * 792 + (r2 - 12) * 12 + dir] = val;
            }
        }
        wave_sync();
    }
}

// ---------------------------------------------------------------------------
// Kernel 2: per-sample small algebra + faithful m = k % n gather + Cholesky.
// ---------------------------------------------------------------------------
__global__ __launch_bounds__(256)
void delan_solve(const float* __restrict__ xu, const float* __restrict__ Lg,
                 const float* __restrict__ gg, const float* __restrict__ yy,
                 const float* __restrict__ dld, const float* __restrict__ dlo,
                 float* __restrict__ out, int n)
{
    const int s = blockIdx.x * blockDim.x + threadIdx.x;
    if (s >= n) return;

    float qd[12], uu[12];
    #pragma unroll
    for (int j = 0; j < 12; ++j) {
        qd[j] = xu[s * 36 + 12 + j];
        uu[j] = xu[s * 36 + 24 + j];
    }

    float L[12][12];
    for (int i = 0; i < 12; ++i)
        for (int j = 0; j < 12; ++j)
            L[i][j] = Lg[(size_t)s * 144 + i * 12 + j];

    // dld_dt / dlo_dt
    float ldt[12];
    #pragma unroll
    for (int l = 0; l < 12; ++l) {
        float a = 0.f;
        for (int j = 0; j < 12; ++j) a += dld[(size_t)s * 144 + l * 12 + j] * qd[j];
        ldt[l] = a;
    }
    float lot[66];
    for (int o = 0; o < 66; ++o) {
        float a = 0.f;
        for (int j = 0; j < 12; ++j) a += dlo[(size_t)s * 792 + o * 12 + j] * qd[j];
        lot[o] = a;
    }

    // z = dL_dt^T qd ; yo = L^T qd (precomputed)
    float z[12], yo[12];
    #pragma unroll
    for (int c = 0; c < 12; ++c) {
        z[c]  = ldt[c] * qd[c];
        yo[c] = yy[s * 12 + c];
    }
    {
        int o = 0;
        for (int r = 1; r < 12; ++r)
            for (int c = 0; c < r; ++c, ++o) z[c] += lot[o] * qd[r];
    }

    // w = dH_dt @ qd = L z + dL_dt yo
    float w[12];
    #pragma unroll
    for (int i = 0; i < 12; ++i) {
        float a = 0.f;
        for (int j = 0; j < 12; ++j) a += L[i][j] * z[j];
        w[i] = a + ldt[i] * yo[i];
    }
    {
        int o = 0;
        for (int r = 1; r < 12; ++r)
            for (int c = 0; c < r; ++c, ++o) w[r] += lot[o] * yo[c];
    }

    // quad with reference's m = (s*12+i) % n gather; quad = 2 * qm^T dLdqi y_m
    float rhs[12];
    for (int i = 0; i < 12; ++i) {
        const int m = (s * 12 + i) % n;
        float ym[12], qm[12];
        #pragma unroll
        for (int j = 0; j < 12; ++j) {
            ym[j] = yy[m * 12 + j];
            qm[j] = xu[m * 36 + 12 + j];
        }
        float t[12];
        #pragma unroll
        for (int r = 0; r < 12; ++r)
            t[r] = dld[(size_t)s * 144 + r * 12 + i] * ym[r];
        const float* lof = dlo + (size_t)s * 792 + i * 66;  // flat reshape view
        {
            int o = 0;
            for (int r = 1; r < 12; ++r)
                for (int c = 0; c < r; ++c, ++o) t[r] += lof[o] * ym[c];
        }
        float q2 = 0.f;
        #pragma unroll
        for (int r = 0; r < 12; ++r) q2 += qm[r] * t[r];
        // c_i = w_i - 0.5*quad_i = w_i - q2
        rhs[i] = uu[i] - (w[i] - q2) - gg[s * 12 + i];
    }

    // H = L L^T + 1e-9 I ; Cholesky (lower, in place)
    float H[12][12];
    for (int i = 0; i < 12; ++i)
        for (int j = 0; j <= i; ++j) {
            float a = 0.f;
            for (int k = 0; k <= j; ++k) a += L[i][k] * L[j][k];
            H[i][j] = a;
        }
    for (int i = 0; i < 12; ++i) H[i][i] += 1e-9f;

    for (int k = 0; k < 12; ++k) {
        float d = sqrtf(H[k][k]);
        H[k][k] = d;
        float inv = 1.f / d;
        for (int i = k + 1; i < 12; ++i) H[i][k] *= inv;
        for (int j = k + 1; j < 12; ++j)
            for (int i = j; i < 12; ++i) H[i][j] -= H[i][k] * H[j][k];
    }
    // solve H x = rhs
    float z2[12], x[12];
    for (int i = 0; i < 12; ++i) {
        float a = rhs[i];
        for (int j = 0; j < i; ++j) a -= H[i][j] * z2[j];
        z2[i] = a / H[i][i];
    }
    for (int i = 11; i >= 0; --i) {
        float a = z2[i];
        for (int j = i + 1; j < 12; ++j) a -= H[j][i] * x[j];
        x[i] = a / H[i][i];
    }

    #pragma unroll
    for (int j = 0; j < 12; ++j) {
        out[s * 36 + j]      = qd[j];
        out[s * 36 + 12 + j] = x[j];
        out[s * 36 + 24 + j] = 0.f;
    }
}

// ---------------------------------------------------------------------------
extern "C" void kernel_launch(void* const* d_in, const int* in_sizes, int n_in,
                              void* d_out, int out_size, void* d_ws, size_t ws_size,
                              hipStream_t stream) {
    const float* xu  = (const float*)d_in[0];
    const float* W1  = (const float*)d_in[1];
    const float* b1  = (const float*)d_in[2];
    const float* W2  = (const float*)d_in[3];
    const float* b2  = (const float*)d_in[4];
    const float* WG  = (const float*)d_in[5];
    const float* bG  = (const float*)d_in[6];
    const float* WLd = (const float*)d_in[7];
    const float* bLd = (const float*)d_in[8];
    const float* WLo = (const float*)d_in[9];
    const float* bLo = (const float*)d_in[10];
    float* out = (float*)d_out;

    const int n = in_sizes[0] / 36;

    float* ws = (float*)d_ws;
    size_t off = 0;
    float* Lg  = ws + off; off += (size_t)n * 144;
    float* gg  = ws + off; off += (size_t)n * 12;
    float* yy  = ws + off; off += (size_t)n * 12;
    float* dld = ws + off; off += (size_t)n * 144;
    float* dlo = ws + off; off += (size_t)n * 792;

    const int blocks1 = (n + SMP_PER_WG - 1) / SMP_PER_WG;
    delan_main<<<blocks1, WPW * 32, SMEM_BYTES, stream>>>(
        xu, W1, b1, W2, b2, WG, bG, WLd, bLd, WLo, bLo,
        Lg, gg, yy, dld, dlo, n);

    const int blocks2 = (n + 255) / 256;
    delan_solve<<<blocks2, 256, 0, stream>>>(xu, Lg, gg, yy, dld, dlo, out, n);
}